// HomoGraphRepresentation_23665269801066
// MI455X (gfx1250) — compile-verified
//
#include <hip/hip_runtime.h>
#include <stdint.h>

#define THREADS 256
#define CPY_UNROLL 4
#define EMPTY_KEY 0xFFFFFFFFFFFFFFFFull
#define NOT_FOUND 0x7FFFFFFF

__device__ __forceinline__ uint64_t mix64(uint64_t x) {
    x ^= x >> 33; x *= 0xff51afd7ed558ccdull;
    x ^= x >> 33; x *= 0xc4ceb9fe1a85ec53ull;
    x ^= x >> 33; return x;
}

// lower_bound on sorted int array (searchsorted semantics of the reference)
__device__ __forceinline__ int lower_bound_i32(const int* __restrict__ a, int n, int v) {
    int lo = 0, hi = n;
    while (lo < hi) { int mid = (lo + hi) >> 1; if (a[mid] < v) lo = mid + 1; else hi = mid; }
    return lo;
}

// ---------------------------------------------------------------------------
// 0) init workspace: last-writer array = -1, hash table keys = EMPTY, min = NOT_FOUND
// ---------------------------------------------------------------------------
__global__ void init_ws_kernel(int* __restrict__ last, int n_nodes,
                               unsigned long long* __restrict__ tkeys,
                               int* __restrict__ tmin, int cap) {
    int total = n_nodes > cap ? n_nodes : cap;
    int stride = gridDim.x * blockDim.x;
    for (int i = blockIdx.x * blockDim.x + threadIdx.x; i < total; i += stride) {
        if (i < n_nodes) last[i] = -1;
        if (i < cap) { tkeys[i] = EMPTY_KEY; tmin[i] = NOT_FOUND; }
    }
}

// ---------------------------------------------------------------------------
// 1) node scatter: effective write order is [src0,dst0,src1,dst1,...]; last wins
// ---------------------------------------------------------------------------
__global__ void scatter_last_kernel(const int* __restrict__ srcID, const int* __restrict__ dstID,
                                    const int* __restrict__ G_idx, int n_nodes, int b,
                                    int* __restrict__ last) {
    int i = blockIdx.x * blockDim.x + threadIdx.x;
    if (i >= b) return;
    int ss = lower_bound_i32(G_idx, n_nodes, srcID[i]);
    int ds = lower_bound_i32(G_idx, n_nodes, dstID[i]);
    atomicMax(&last[ss], 2 * i);        // src written first
    atomicMax(&last[ds], 2 * i + 1);    // dst written second (wins ties)
}

// ---------------------------------------------------------------------------
// 2) insert the B query keys (src_slot*N + dst_slot) into the hash table
// ---------------------------------------------------------------------------
__global__ void qinsert_kernel(const int* __restrict__ srcID, const int* __restrict__ dstID,
                               const int* __restrict__ G_idx, int n_nodes, int b,
                               unsigned long long* __restrict__ tkeys, unsigned cap_mask) {
    int i = blockIdx.x * blockDim.x + threadIdx.x;
    if (i >= b) return;
    int ss = lower_bound_i32(G_idx, n_nodes, srcID[i]);
    int ds = lower_bound_i32(G_idx, n_nodes, dstID[i]);
    unsigned long long key = (unsigned long long)ss * (unsigned long long)n_nodes
                           + (unsigned long long)ds;
    unsigned h = (unsigned)mix64(key) & cap_mask;
    for (;;) {
        unsigned long long prev = atomicCAS(&tkeys[h], EMPTY_KEY, key);
        if (prev == EMPTY_KEY || prev == key) break;
        h = (h + 1) & cap_mask;
    }
}

// ---------------------------------------------------------------------------
// 3) scan all E edges once; atomicMin edge index into matching query slot
//    (reference: stable argsort + leftmost searchsorted == min matching index)
// ---------------------------------------------------------------------------
__global__ void edge_scan_kernel(const int* __restrict__ ei0, const int* __restrict__ ei1,
                                 long long n_edges, long long n_nodes,
                                 const unsigned long long* __restrict__ tkeys,
                                 int* __restrict__ tmin, unsigned cap_mask) {
    long long stride = (long long)gridDim.x * blockDim.x;
    for (long long j = (long long)blockIdx.x * blockDim.x + threadIdx.x; j < n_edges; j += stride) {
        unsigned long long key = (unsigned long long)ei0[j] * (unsigned long long)n_nodes
                               + (unsigned long long)ei1[j];
        unsigned h = (unsigned)mix64(key) & cap_mask;
        for (;;) {
            unsigned long long cur = tkeys[h];
            if (cur == EMPTY_KEY) break;                 // no query with this key
            if (cur == key) { atomicMin(&tmin[h], (int)j); break; }
            h = (h + 1) & cap_mask;
        }
    }
}

// ---------------------------------------------------------------------------
// 4) node section of output + updated flags
// ---------------------------------------------------------------------------
__global__ void node_out_kernel(const float* __restrict__ G_x,
                                const float* __restrict__ src_f, const float* __restrict__ dst_f,
                                const int* __restrict__ last, int n_nodes, int feat,
                                float* __restrict__ out_nodes, float* __restrict__ out_flags) {
    long long total = (long long)n_nodes * feat;
    long long stride = (long long)gridDim.x * blockDim.x;
    for (long long i = (long long)blockIdx.x * blockDim.x + threadIdx.x; i < total; i += stride) {
        int n = (int)(i / feat);
        int f = (int)(i - (long long)n * feat);
        int w = last[n];                                  // L2-resident, reused 15x
        float v;
        if (w >= 0) {
            long long row = (long long)(w >> 1);
            v = (w & 1) ? dst_f[row * feat + f] : src_f[row * feat + f];
        } else {
            v = G_x[i];
        }
        out_nodes[i] = v;
        if (f == 0) out_flags[n] = (w >= 0) ? 1.0f : 0.0f;
    }
}

// ---------------------------------------------------------------------------
// 5) bulk edge_attr passthrough copy (240 MB) via CDNA5 async global<->LDS
//    B128 path, 4 transfers in flight per lane between ASYNCcnt drains.
//    NT hints keep the one-touch stream out of the 192 MB L2 so the reused
//    data (edge_index 2nd pass, hash table, last[]) stays resident.
// ---------------------------------------------------------------------------
__global__ void stream_copy_b128_async_kernel(const float4* __restrict__ src,
                                              float4* __restrict__ dst, long long n4) {
    __shared__ float4 stage[CPY_UNROLL * THREADS];
    unsigned lds0 = (unsigned)(uintptr_t)&stage[threadIdx.x];
    long long block_span = (long long)blockDim.x * CPY_UNROLL;
    long long grid_span  = (long long)gridDim.x * block_span;
    for (long long i = (long long)blockIdx.x * block_span + threadIdx.x; i < n4; i += grid_span) {
#pragma unroll
        for (int u = 0; u < CPY_UNROLL; ++u) {
            long long idx = i + (long long)u * blockDim.x;
            if (idx < n4) {
                unsigned long long gsrc = (unsigned long long)(uintptr_t)(src + idx);
                unsigned lds = lds0 + (unsigned)(u * THREADS * sizeof(float4));
                asm volatile("global_load_async_to_lds_b128 %0, %1, off th:TH_LOAD_NT"
                             :: "v"(lds), "v"(gsrc) : "memory");
            }
        }
        asm volatile("s_wait_asynccnt 0" ::: "memory");
#pragma unroll
        for (int u = 0; u < CPY_UNROLL; ++u) {
            long long idx = i + (long long)u * blockDim.x;
            if (idx < n4) {
                unsigned long long gdst = (unsigned long long)(uintptr_t)(dst + idx);
                unsigned lds = lds0 + (unsigned)(u * THREADS * sizeof(float4));
                asm volatile("global_store_async_from_lds_b128 %0, %1, off th:TH_STORE_NT"
                             :: "v"(gdst), "v"(lds) : "memory");
            }
        }
        asm volatile("s_wait_asynccnt 0" ::: "memory");
    }
}

// scalar tail for copies not divisible by 16 bytes (unused for F=15,E=4M but generic)
__global__ void copy_tail_kernel(const float* __restrict__ src, float* __restrict__ dst,
                                 long long start, long long total) {
    long long stride = (long long)gridDim.x * blockDim.x;
    for (long long i = start + (long long)blockIdx.x * blockDim.x + threadIdx.x; i < total; i += stride)
        dst[i] = src[i];
}

// ---------------------------------------------------------------------------
// 6) overwrite found edge rows with edge_feature (drop if not found)
// ---------------------------------------------------------------------------
__global__ void apply_edge_kernel(const int* __restrict__ srcID, const int* __restrict__ dstID,
                                  const int* __restrict__ G_idx, int n_nodes, int b, int feat,
                                  const float* __restrict__ edge_feature,
                                  const unsigned long long* __restrict__ tkeys,
                                  const int* __restrict__ tmin, unsigned cap_mask,
                                  float* __restrict__ out_edges) {
    int q = blockIdx.x * blockDim.x + threadIdx.x;
    if (q >= b) return;
    int ss = lower_bound_i32(G_idx, n_nodes, srcID[q]);
    int ds = lower_bound_i32(G_idx, n_nodes, dstID[q]);
    unsigned long long key = (unsigned long long)ss * (unsigned long long)n_nodes
                           + (unsigned long long)ds;
    unsigned h = (unsigned)mix64(key) & cap_mask;
    int e = NOT_FOUND;
    for (;;) {
        unsigned long long cur = tkeys[h];
        if (cur == EMPTY_KEY) break;
        if (cur == key) { e = tmin[h]; break; }
        h = (h + 1) & cap_mask;
    }
    if (e == NOT_FOUND) return;  // query edge not present -> dropped
    float* drow = out_edges + (long long)e * feat;
    const float* srow = edge_feature + (long long)q * feat;
    for (int f = 0; f < feat; ++f) drow[f] = srow[f];
}

// ---------------------------------------------------------------------------
// 7) edge_index -> float tail of output (values < 2^18, exact in f32);
//    vectorized int4->float4; second read of edge_index is L2-resident.
// ---------------------------------------------------------------------------
__global__ void eidx_out_kernel(const int* __restrict__ edge_index, long long n2,
                                float* __restrict__ out) {
    long long n4 = n2 >> 2;
    long long stride = (long long)gridDim.x * blockDim.x;
    const int4* src4 = (const int4*)edge_index;
    float4* dst4 = (float4*)out;
    for (long long i = (long long)blockIdx.x * blockDim.x + threadIdx.x; i < n4; i += stride) {
        int4 v = src4[i];
        float4 f;
        f.x = (float)v.x; f.y = (float)v.y; f.z = (float)v.z; f.w = (float)v.w;
        dst4[i] = f;
    }
    if (blockIdx.x == 0 && (long long)threadIdx.x < (n2 & 3)) {
        long long t = (n4 << 2) + threadIdx.x;
        out[t] = (float)edge_index[t];
    }
}

static inline int grid_for(long long n, int cap_blocks) {
    long long b = (n + THREADS - 1) / THREADS;
    if (b > cap_blocks) b = cap_blocks;
    if (b < 1) b = 1;
    return (int)b;
}

extern "C" void kernel_launch(void* const* d_in, const int* in_sizes, int n_in,
                              void* d_out, int out_size, void* d_ws, size_t ws_size,
                              hipStream_t stream) {
    const float* G_x         = (const float*)d_in[0];
    const int*   G_idx       = (const int*)d_in[1];
    const int*   edge_index  = (const int*)d_in[2];
    const float* edge_attr   = (const float*)d_in[3];
    const int*   srcID       = (const int*)d_in[4];
    const int*   dstID       = (const int*)d_in[5];
    const float* src_feature = (const float*)d_in[6];
    const float* dst_feature = (const float*)d_in[7];
    const float* edge_feature= (const float*)d_in[8];
    (void)n_in; (void)out_size; (void)ws_size;

    const int       n_nodes = in_sizes[1];
    const int       feat    = in_sizes[0] / n_nodes;
    const long long n_edges = (long long)in_sizes[2] / 2;
    const int       b       = in_sizes[4];

    // hash capacity: power of two, load factor <= 25%
    int cap = 1 << 10;
    while (cap < 4 * b) cap <<= 1;
    const unsigned cap_mask = (unsigned)(cap - 1);

    // workspace layout (~2.6 MB total)
    char* ws = (char*)d_ws;
    int* last = (int*)ws;
    size_t off = ((size_t)n_nodes * sizeof(int) + 255) & ~(size_t)255;
    unsigned long long* tkeys = (unsigned long long*)(ws + off);
    off += (size_t)cap * sizeof(unsigned long long);
    int* tmin = (int*)(ws + off);

    // output layout: [x_new | edge_attr_new | updated | edge_index(float)]
    float* out       = (float*)d_out;
    float* out_nodes = out;
    float* out_edges = out + (long long)n_nodes * feat;
    float* out_flags = out_edges + n_edges * feat;
    float* out_eidx  = out_flags + n_nodes;

    const int* ei0 = edge_index;
    const int* ei1 = edge_index + n_edges;

    // 0) init scratch
    {
        long long n = n_nodes > cap ? n_nodes : cap;
        init_ws_kernel<<<grid_for(n, 2048), THREADS, 0, stream>>>(last, n_nodes, tkeys, tmin, cap);
    }
    // 1) node last-writer scatter
    scatter_last_kernel<<<grid_for(b, 1 << 20), THREADS, 0, stream>>>(
        srcID, dstID, G_idx, n_nodes, b, last);
    // 2) hash the queries
    qinsert_kernel<<<grid_for(b, 1 << 20), THREADS, 0, stream>>>(
        srcID, dstID, G_idx, n_nodes, b, tkeys, cap_mask);
    // 3) one streaming pass over all edges
    edge_scan_kernel<<<grid_for(n_edges, 4096), THREADS, 0, stream>>>(
        ei0, ei1, n_edges, (long long)n_nodes, tkeys, tmin, cap_mask);
    // 4) node output + flags
    node_out_kernel<<<grid_for((long long)n_nodes * feat, 4096), THREADS, 0, stream>>>(
        G_x, src_feature, dst_feature, last, n_nodes, feat, out_nodes, out_flags);
    // 5) bulk edge_attr copy via async global<->LDS B128 streaming (NT)
    {
        long long total = n_edges * feat;
        long long n4 = total >> 2;           // 16B chunks
        if (n4 > 0) {
            long long blocks = (n4 + (long long)THREADS * CPY_UNROLL - 1)
                             / ((long long)THREADS * CPY_UNROLL);
            if (blocks > 8192) blocks = 8192;
            if (blocks < 1) blocks = 1;
            stream_copy_b128_async_kernel<<<(int)blocks, THREADS, 0, stream>>>(
                (const float4*)edge_attr, (float4*)out_edges, n4);
        }
        if (total & 3)
            copy_tail_kernel<<<1, THREADS, 0, stream>>>(edge_attr, out_edges, n4 << 2, total);
    }
    // 6) scatter updated edge rows
    apply_edge_kernel<<<grid_for(b, 1 << 20), THREADS, 0, stream>>>(
        srcID, dstID, G_idx, n_nodes, b, feat, edge_feature, tkeys, tmin, cap_mask, out_edges);
    // 7) edge_index as float
    eidx_out_kernel<<<grid_for((2 * n_edges) >> 2, 4096), THREADS, 0, stream>>>(
        edge_index, 2 * n_edges, out_eidx);
}